// GNN_9088150798514
// MI455X (gfx1250) — compile-verified
//
#include <hip/hip_runtime.h>

typedef __attribute__((ext_vector_type(2))) float v2f;
typedef __attribute__((ext_vector_type(8))) float v8f;

#define DIMF 64
#define LDSP 68    // A-tile LDS row stride (floats): 272B, 16B-aligned, conflict-free
#define LDWP 160   // W pair-row stride (floats): pairs p,p+1 on disjoint bank halves

// out[M,64] (= or +=) f(A)[M,64] @ W[64,64] + bias
// f(A) = A (+A2 if ADD_A2), or relu(A*bnscale+bnshift) if BN_RELU.
template <bool ADD_A2, bool BN_RELU, bool ACCUM>
__global__ __launch_bounds__(256) void gemm64_kernel(
    const float* __restrict__ A, const float* __restrict__ A2,
    const float* __restrict__ W, const float* __restrict__ bias,
    const float* __restrict__ bnscale, const float* __restrict__ bnshift,
    float* __restrict__ out, int M) {
  __shared__ float W_s[32 * LDWP];      // pair-interleaved: {W[2p][n],W[2p+1][n]}
  __shared__ float A_s[8][16 * LDSP];
  const int tid = threadIdx.x;
  const int lane = tid & 31;
  const int w = tid >> 5;

  // Stage W (64x64 row-major, K x N) into pair-interleaved LDS.
#pragma unroll
  for (int i = 0; i < 16; ++i) {
    int idx = tid + i * 256;
    int k = idx >> 6, n = idx & 63;
    W_s[(k >> 1) * LDWP + (n << 1) + (k & 1)] = W[idx];
  }

  const int mt = blockIdx.x * 8 + w;
  const int row0 = mt * 16;
  const bool active = (row0 < M);  // M is a multiple of 16 (100000 = 16*6250)

  if (active) {
    float* As = &A_s[w][0];
#pragma unroll
    for (int j = 0; j < 8; ++j) {
      int flat = j * 128 + lane * 4;
      int r = flat >> 6, c = flat & 63;
      float4 v = *(const float4*)&A[(size_t)(row0 + r) * DIMF + c];
      if (ADD_A2) {
        float4 u = *(const float4*)&A2[(size_t)(row0 + r) * DIMF + c];
        v.x += u.x; v.y += u.y; v.z += u.z; v.w += u.w;
      }
      if (BN_RELU) {
        float4 s = *(const float4*)&bnscale[c];
        float4 t = *(const float4*)&bnshift[c];
        v.x = fmaxf(fmaf(v.x, s.x, t.x), 0.f);
        v.y = fmaxf(fmaf(v.y, s.y, t.y), 0.f);
        v.z = fmaxf(fmaf(v.z, s.z, t.z), 0.f);
        v.w = fmaxf(fmaf(v.w, s.w, t.w), 0.f);
      }
      *(float4*)&As[r * LDSP + c] = v;
    }
  }
  __syncthreads();
  if (!active) return;

  const float* As = &A_s[w][0];
  const int hi = lane >> 4;   // half-wave select
  const int lq = lane & 15;   // M (for A) / N (for B) within tile
  v8f acc[4] = {};

#pragma unroll
  for (int k0 = 0; k0 < 16; ++k0) {
    const int ka = k0 * 4 + 2 * hi;
    // A fragment: lane holds A[M=lq][ka], A[M=lq][ka+1]  (ISA 16x4 f32 layout)
    v2f a = *(const v2f*)&As[lq * LDSP + ka];
    // B fragment: {B[ka][col], B[ka+1][col]} = one b64 from pair row p.
    const int p = k0 * 2 + hi;
    const float* wp = &W_s[p * LDWP + lq * 2];
#pragma unroll
    for (int n = 0; n < 4; ++n) {
      v2f b = *(const v2f*)&wp[n * 32];
      acc[n] = __builtin_amdgcn_wmma_f32_16x16x4_f32(
          false, a, false, b, (short)0, acc[n], false, false);
    }
  }

  // Epilogue: C/D layout — VGPR v: lanes0-15 -> M=v, lanes16-31 -> M=v+8; N=lq.
#pragma unroll
  for (int n = 0; n < 4; ++n) {
    int col = n * 16 + lq;
    float bv = bias[col];
#pragma unroll
    for (int v = 0; v < 8; ++v) {
      int row = row0 + v + hi * 8;
      float val = acc[n][v] + bv;
      size_t o = (size_t)row * DIMF + col;
      if (ACCUM) out[o] += val; else out[o] = val;
    }
  }
}

// agg[dst] += x[src] for edges of relation `rel`
__global__ __launch_bounds__(256) void scatter_kernel(
    const float* __restrict__ x, const int* __restrict__ src,
    const int* __restrict__ dst, const int* __restrict__ et, int rel, int E,
    float* __restrict__ agg) {
  int total = E * DIMF;
  for (int i = blockIdx.x * blockDim.x + threadIdx.x; i < total;
       i += gridDim.x * blockDim.x) {
    int e = i >> 6, f = i & 63;
    if (et[e] == rel)
      atomicAdd(&agg[(size_t)dst[e] * DIMF + f],
                x[(size_t)src[e] * DIMF + f]);
  }
}

// per-column sum and sum of squares -> stats[0:64]=sum, stats[64:128]=sumsq
__global__ __launch_bounds__(256) void colstats_kernel(
    const float* __restrict__ t, int M, float* __restrict__ stats) {
  __shared__ float sred[256], qred[256];
  int col = threadIdx.x & 63;
  int rg = threadIdx.x >> 6;
  float s = 0.f, q = 0.f;
  for (int row = blockIdx.x * 4 + rg; row < M; row += gridDim.x * 4) {
    float v = t[(size_t)row * DIMF + col];
    s += v; q += v * v;
  }
  sred[threadIdx.x] = s;
  qred[threadIdx.x] = q;
  __syncthreads();
  if (rg == 0) {
    s = sred[col] + sred[col + 64] + sred[col + 128] + sred[col + 192];
    q = qred[col] + qred[col + 64] + qred[col + 128] + qred[col + 192];
    atomicAdd(&stats[col], s);
    atomicAdd(&stats[64 + col], q);
  }
}

__global__ void bnfinal_kernel(const float* __restrict__ stats,
                               const float* __restrict__ gamma,
                               const float* __restrict__ beta, float invM,
                               float* __restrict__ scale,
                               float* __restrict__ shift) {
  int c = threadIdx.x;  // 64 threads
  float mean = stats[c] * invM;
  float var = stats[64 + c] * invM - mean * mean;   // biased var, matches ref
  float rstd = rsqrtf(var + 1e-5f);
  float sc = rstd * gamma[c];
  scale[c] = sc;
  shift[c] = beta[c] - mean * sc;
}

__global__ __launch_bounds__(256) void relu_kernel(const float* __restrict__ in,
                                                   float* __restrict__ out,
                                                   int n) {
  int i = blockIdx.x * blockDim.x + threadIdx.x;
  if (i < n) out[i] = fmaxf(in[i], 0.f);
}

__global__ __launch_bounds__(256) void pool_scatter_kernel(
    const float* __restrict__ x, const int* __restrict__ batch,
    float* __restrict__ out, float* __restrict__ cnt, int Nn) {
  int total = Nn * DIMF;
  for (int i = blockIdx.x * blockDim.x + threadIdx.x; i < total;
       i += gridDim.x * blockDim.x) {
    int node = i >> 6, f = i & 63;
    int g = batch[node];
    atomicAdd(&out[(size_t)g * DIMF + f], x[i]);
    if (f == 0) atomicAdd(&cnt[g], 1.f);
  }
}

__global__ __launch_bounds__(256) void pool_div_kernel(float* __restrict__ out,
                                                       const float* __restrict__ cnt,
                                                       int total) {
  int i = blockIdx.x * blockDim.x + threadIdx.x;
  if (i < total) out[i] /= fmaxf(cnt[i >> 6], 1.f);
}

extern "C" void kernel_launch(void* const* d_in, const int* in_sizes, int n_in,
                              void* d_out, int out_size, void* d_ws,
                              size_t ws_size, hipStream_t stream) {
  const float* x     = (const float*)d_in[0];
  const int*   eidx  = (const int*)d_in[1];
  const int*   etype = (const int*)d_in[2];
  const int*   batch = (const int*)d_in[3];
  const float* Wsl   = (const float*)d_in[4];
  const float* bsl   = (const float*)d_in[5];
  const float* W1    = (const float*)d_in[6];
  const float* b1    = (const float*)d_in[7];
  const float* gamma = (const float*)d_in[8];
  const float* beta  = (const float*)d_in[9];
  const float* W2    = (const float*)d_in[10];
  const float* b2    = (const float*)d_in[11];

  const int E  = in_sizes[2];
  const int Nn = in_sizes[0] / DIMF;
  const int* src = eidx;
  const int* dst = eidx + E;

  float* ws = (float*)d_ws;
  const size_t nfeat = (size_t)Nn * DIMF;
  float* x_cur   = ws;
  float* x_acc   = x_cur + nfeat;
  float* agg     = x_acc + nfeat;
  float* t1      = agg + nfeat;
  float* stats   = t1 + nfeat;      // 128
  float* bnscale = stats + 128;     // 64
  float* bnshift = bnscale + 64;    // 64
  float* cnt     = bnshift + 64;    // 128

  hipMemcpyAsync(x_cur, x, nfeat * sizeof(float), hipMemcpyDeviceToDevice,
                 stream);

  const int mtiles = (Nn + 15) / 16;
  const int gemm_blocks = (mtiles + 7) / 8;
  dim3 B(256);

  for (int l = 0; l < 3; ++l) {
    // self-loop linear: x_acc = x_cur @ Wsl[l] + bsl[l]
    gemm64_kernel<false, false, false><<<gemm_blocks, B, 0, stream>>>(
        x_cur, nullptr, Wsl + (size_t)l * DIMF * DIMF, bsl + (size_t)l * DIMF,
        nullptr, nullptr, x_acc, Nn);
    for (int r = 0; r < 4; ++r) {
      const size_t off = (size_t)l * 4 + r;
      hipMemsetAsync(agg, 0, nfeat * sizeof(float), stream);
      scatter_kernel<<<2048, B, 0, stream>>>(x_cur, src, dst, etype, r, E, agg);
      // t1 = (x_cur + agg) @ W1[l,r] + b1[l,r]
      gemm64_kernel<true, false, false><<<gemm_blocks, B, 0, stream>>>(
          x_cur, agg, W1 + off * DIMF * DIMF, b1 + off * DIMF, nullptr,
          nullptr, t1, Nn);
      hipMemsetAsync(stats, 0, 128 * sizeof(float), stream);
      colstats_kernel<<<512, B, 0, stream>>>(t1, Nn, stats);
      bnfinal_kernel<<<1, 64, 0, stream>>>(stats, gamma + off * DIMF,
                                           beta + off * DIMF, 1.0f / (float)Nn,
                                           bnscale, bnshift);
      // x_acc += relu(BN(t1)) @ W2[l,r] + b2[l,r]
      gemm64_kernel<false, true, true><<<gemm_blocks, B, 0, stream>>>(
          t1, nullptr, W2 + off * DIMF * DIMF, b2 + off * DIMF, bnscale,
          bnshift, x_acc, Nn);
    }
    if (l != 2)
      relu_kernel<<<(int)((nfeat + 255) / 256), B, 0, stream>>>(x_acc, x_cur,
                                                                (int)nfeat);
  }

  hipMemsetAsync(d_out, 0, (size_t)out_size * sizeof(float), stream);
  hipMemsetAsync(cnt, 0, 128 * sizeof(float), stream);
  pool_scatter_kernel<<<4096, B, 0, stream>>>(x_acc, batch, (float*)d_out, cnt,
                                              Nn);
  pool_div_kernel<<<(out_size + 255) / 256, B, 0, stream>>>((float*)d_out, cnt,
                                                            out_size);
}